// InterventionAwareStructure_82240033784129
// MI455X (gfx1250) — compile-verified
//
#include <hip/hip_runtime.h>

typedef float v2f __attribute__((ext_vector_type(2)));
typedef float v8f __attribute__((ext_vector_type(8)));

#define DDIM   512
#define NREG   16
#define BATCH  256
#define CTILES (DDIM / 16)          // 32 column tiles of 16
#define NTILES (BATCH * CTILES)     // 8192 partials

// One wave (32 threads) per (batch, column-tile). Column sums via
// V_WMMA_F32_16X16X4_F32 with an all-ones A operand:
//   D[m][n] += sum_k B[k][n]   (B = 4x16 tile of A_per_env)
// Result is invariant to K->lane-half ordering inside the B operand since
// A is all ones; only N->lane mapping (lanes 0-15 = N 0-15) matters.
__global__ __launch_bounds__(32)
void colsum_wmma_kernel(const float* __restrict__ A,
                        const float* __restrict__ mask,
                        const int*   __restrict__ regimes,
                        float*       __restrict__ partial) {
    const int tile = blockIdx.x;
    const int b    = tile >> 5;          // / CTILES
    const int ct   = tile & (CTILES - 1);
    const int c0   = ct * 16;
    const int lane = threadIdx.x;        // 0..31 (wave32)

    const float* Ab    = A + (size_t)b * DDIM * DDIM;
    const int    col   = c0 + (lane & 15);
    const int    rowSel = (lane >> 4) * 2;   // lanes 0-15 -> rows k0,k0+1 ; lanes 16-31 -> k0+2,k0+3

    v2f ones; ones.x = 1.0f; ones.y = 1.0f;
    v8f acc = {};

    #pragma unroll 4
    for (int k0 = 0; k0 < DDIM; k0 += 4) {
        const float* p = Ab + (size_t)(k0 + rowSel) * DDIM + col;
        v2f bmat;
        bmat.x = p[0];       // row k0 + rowSel
        bmat.y = p[DDIM];    // row k0 + rowSel + 1
        // 8 args: (neg_a, A, neg_b, B, c_mod, C, reuse_a, reuse_b)
        acc = __builtin_amdgcn_wmma_f32_16x16x4_f32(
            false, ones, false, bmat, (short)0, acc, false, false);
    }

    // acc[0]: lanes 0-15 hold M=0 row, lanes 16-31 hold M=8 row; all rows of D
    // are identical (A was all ones), so acc[0] == colsum of column `col`.
    float colsum = acc[0];

    const int e     = regimes[b];
    const int valid = (e < NREG) ? 1 : 0;            // reference: valid = regimes < n_regimes
    int ec = e < 0 ? 0 : (e > NREG - 1 ? NREG - 1 : e);
    float mval = valid ? mask[(size_t)ec * DDIM + col] : 0.0f;

    float v = (lane < 16) ? colsum * mval : 0.0f;    // avoid double-counting upper half
    #pragma unroll
    for (int off = 16; off > 0; off >>= 1)
        v += __shfl_xor(v, off, 32);

    if (lane == 0) partial[tile] = v;                // fully rewritten every launch
}

// Deterministic fixed-order reduction of the 8192 partials + mask count.
__global__ __launch_bounds__(256)
void finalize_kernel(const float* __restrict__ partial,
                     const float* __restrict__ mask,
                     const int*   __restrict__ regimes,
                     float*       __restrict__ out) {
    __shared__ float sN[256];
    __shared__ float sC[256];
    const int t = threadIdx.x;

    float n = 0.0f;
    for (int i = t; i < NTILES; i += 256)
        n += partial[i];

    // count contribution of batch b = t  (BATCH == 256)
    float c = 0.0f;
    const int e = regimes[t];
    if (e < NREG) {
        int ec = e < 0 ? 0 : e;
        const float* mr = mask + (size_t)ec * DDIM;
        for (int j = 0; j < DDIM; ++j)
            c += mr[j];
    }

    sN[t] = n;
    sC[t] = c;
    __syncthreads();

    #pragma unroll
    for (int s = 128; s > 0; s >>= 1) {
        if (t < s) { sN[t] += sN[t + s]; sC[t] += sC[t + s]; }
        __syncthreads();
    }

    if (t == 0) {
        const float numer = sN[0];
        const float count = sC[0];
        const float loss  = (count > 0.0f) ? numer / count : numer;
        out[0] = 1.0f * loss;   // INTERVENTION_STRENGTH == 1.0
    }
}

extern "C" void kernel_launch(void* const* d_in, const int* in_sizes, int n_in,
                              void* d_out, int out_size, void* d_ws, size_t ws_size,
                              hipStream_t stream) {
    const float* A       = (const float*)d_in[0];   // [256, 512, 512] f32
    const float* mask    = (const float*)d_in[1];   // [16, 512] f32
    const int*   regimes = (const int*)d_in[2];     // [256] int
    float*       out     = (float*)d_out;           // scalar f32
    float*       partial = (float*)d_ws;            // NTILES floats (32 KB)

    colsum_wmma_kernel<<<NTILES, 32, 0, stream>>>(A, mask, regimes, partial);
    finalize_kernel<<<1, 256, 0, stream>>>(partial, mask, regimes, out);
}